// TargetGenerator_2482491097553
// MI455X (gfx1250) — compile-verified
//
#include <hip/hip_runtime.h>

#define DEV __device__ __forceinline__

constexpr int B = 16;
constexpr int A = 22500;
constexpr int G = 50;
constexpr int NS = 256;               // N_SAMPLE
constexpr float POS_THR = 0.7f;
constexpr float IMGH = 800.f, IMGW = 800.f;
constexpr unsigned HALF = (unsigned)(B * A) / 2u;   // threefry counter split
constexpr int NBIN = 1024;
constexpr int BCAP = 1024;

typedef __attribute__((ext_vector_type(4))) unsigned int v4u;
typedef __attribute__((ext_vector_type(8))) unsigned int v8u;

// ---------------------------------------------------------------- CDNA5 bits
DEV void wait_tensorcnt0() {
#if __has_builtin(__builtin_amdgcn_s_wait_tensorcnt)
  __builtin_amdgcn_s_wait_tensorcnt(0);
#else
  asm volatile("s_wait_tensorcnt 0x0" ::: "memory");
#endif
}

// byte offset of a __shared__ object within the workgroup LDS allocation
DEV unsigned lds_offset_of(const void* p) {
  return (unsigned)(unsigned long long)(__attribute__((address_space(3))) const char*)p;
}

// TDM: DMA `ndw` dwords (contiguous) from global `src` into LDS at `lds_off`.
// D# packed per CDNA5 ISA ch.8: 1-D tile (tile_dim0 = ndw, tile_dim1 = 1).
DEV void tdm_load_dwords_to_lds(const void* src, unsigned lds_off, unsigned ndw) {
  unsigned long long ga = (unsigned long long)src;
  v4u g0 = { 1u,                                          // count=1, user mode
             lds_off,                                     // lds_addr
             (unsigned)ga,                                // global_addr[31:0]
             (unsigned)((ga >> 32) & 0x01FFFFFFull)       // global_addr[56:32]
                 | 0x80000000u };                         // type=2 ("image")
  v8u g1 = { 0x00020000u,                                 // data_size=2 (4B), mask=0
             (ndw & 0xFFFFu) << 16,                       // tensor_dim0[15:0]
             ((ndw >> 16) & 0xFFFFu) | (1u << 16),        // dim0[31:16] | dim1 lo = 1
             (ndw & 0xFFFFu) << 16,                       // dim1 hi = 0 | tile_dim0
             1u,                                          // tile_dim1=1, tile_dim2=0
             ndw,                                         // dim0_stride[31:0]
             (ndw & 0xFFFFu) << 16,                       // stride0 hi=0 | stride1 lo
             (ndw >> 16) };                               // stride1 hi
  asm volatile("tensor_load_to_lds %0, %1" :: "s"(g0), "s"(g1) : "memory");
}

// ---------------------------------------------------------------- JAX PRNG
DEV unsigned rotl32(unsigned x, int r) { return (x << r) | (x >> (32 - r)); }

// Reproduces jax.random.uniform(jax.random.key(42), (B, A)) element m=b*A+a.
// threefry-2x32, key=(0,42), counters split iota(N) at N/2.
DEV float jax_uniform01(unsigned m) {
  const unsigned k = (m < HALF) ? m : m - HALF;
  unsigned x0 = k, x1 = k + HALF;
  const unsigned ks0 = 0u, ks1 = 42u, ks2 = 0x1BD11BDAu ^ ks0 ^ ks1;
  const unsigned ks[3] = {ks0, ks1, ks2};
  const int rot[8] = {13, 15, 26, 6, 17, 29, 16, 24};
  x0 += ks0; x1 += ks1;
#pragma unroll
  for (int i = 0; i < 5; ++i) {
#pragma unroll
    for (int j = 0; j < 4; ++j) {
      x0 += x1; x1 = rotl32(x1, rot[(i & 1) * 4 + j]); x1 ^= x0;
    }
    x0 += ks[(i + 1) % 3];
    x1 += ks[(i + 2) % 3] + (unsigned)(i + 1);
  }
  const unsigned bits = (m < HALF) ? x0 : x1;
  return __uint_as_float((bits >> 9) | 0x3F800000u) - 1.0f;
}

DEV int bin_of(float r) {
  int v = (int)(r * (float)NBIN);
  return v < 0 ? 0 : (v > NBIN - 1 ? NBIN - 1 : v);
}

// ---------------------------------------------------------------- kernels
__global__ void k0_init(unsigned long long* __restrict__ best) {
  const int i = blockIdx.x * blockDim.x + threadIdx.x;
  if (i < B * G) best[i] = 0ull;
}

__global__ __launch_bounds__(256) void k1_iou(
    const float* __restrict__ anchors, const float* __restrict__ gt,
    int* __restrict__ gt_idx_ws, float* __restrict__ r_ws,
    float* __restrict__ lab_ws, unsigned long long* __restrict__ best_ws) {
  const int b = blockIdx.y;
  __shared__ float sgt[G * 4];
  __shared__ unsigned long long sbest[G];

  if (threadIdx.x < G) sbest[threadIdx.x] = 0ull;
  if (threadIdx.x < 32) {                 // wave 0 drives the TDM DMA
    tdm_load_dwords_to_lds(gt + (size_t)b * G * 4, lds_offset_of(sgt), G * 4);
    wait_tensorcnt0();
  }
  __syncthreads();

  const int a = blockIdx.x * blockDim.x + threadIdx.x;
  const bool valid = a < A;
  float ay1 = 0.f, ax1 = 0.f, ay2 = 0.f, ax2 = 0.f, area_a = 0.f;
  bool inside = false;
  if (valid) {
    const float4 av = ((const float4*)anchors)[(size_t)b * A + a];
    ay1 = av.x; ax1 = av.y; ay2 = av.z; ax2 = av.w;
    inside = (ay1 >= 0.f) && (ax1 >= 0.f) && (ay2 <= IMGH) && (ax2 <= IMGW);
    area_a = (ay2 - ay1) * (ax2 - ax1);
  }

  float best_iou = -1.f;
  int best_g = 0;
  for (int g = 0; g < G; ++g) {
    float iou = 0.f;
    if (valid) {
      const float gy1 = sgt[g * 4 + 0], gx1 = sgt[g * 4 + 1];
      const float gy2 = sgt[g * 4 + 2], gx2 = sgt[g * 4 + 3];
      const float ih = fmaxf(fminf(ay2, gy2) - fmaxf(ay1, gy1), 0.f);
      const float iw = fmaxf(fminf(ax2, gx2) - fmaxf(ax1, gx1), 0.f);
      const float inter = ih * iw;
      const float area_g = (gy2 - gy1) * (gx2 - gx1);
      iou = inter / fmaxf(area_a + area_g - inter, 1e-9f);
      iou = inside ? iou : 0.f;
      if (iou > best_iou) { best_iou = iou; best_g = g; }   // first-max tie-break
    }
    // key = (iou_bits, ~anchor): max() picks highest IoU, then lowest index
    unsigned long long key =
        valid ? (((unsigned long long)__float_as_uint(iou) << 32) |
                 (unsigned long long)(0xFFFFFFFFu - (unsigned)a))
              : 0ull;
    for (int off = 16; off > 0; off >>= 1) {                // wave32 reduce
      unsigned long long o = __shfl_down(key, off, 32);
      key = o > key ? o : key;
    }
    if ((threadIdx.x & 31) == 0) atomicMax(&sbest[g], key); // ds_max_u64
  }

  if (valid) {
    const size_t ba = (size_t)b * A + a;
    gt_idx_ws[ba] = best_g;
    lab_ws[ba] = inside ? (best_iou >= POS_THR ? 1.f : 0.f) : -1.f;
    r_ws[ba] = jax_uniform01((unsigned)ba);
  }
  __syncthreads();
  if (threadIdx.x < G)
    atomicMax(&best_ws[b * G + threadIdx.x], sbest[threadIdx.x]);
}

__global__ void k2_best(const unsigned long long* __restrict__ best,
                        float* __restrict__ lab_ws) {
  const int i = blockIdx.x * blockDim.x + threadIdx.x;
  if (i >= B * G) return;
  const int b = i / G;
  const unsigned a = 0xFFFFFFFFu - (unsigned)(best[i] & 0xFFFFFFFFull);
  if (a < (unsigned)A) {
    float* lp = lab_ws + (size_t)b * A + a;
    if (*lp >= 0.f) *lp = 1.f;   // set 1 unless overwritten by !inside (-1)
  }
}

// Per-batch exact k-smallest-r sampling (histogram select + boundary ranking).
// Reads lab_ws (never written here) -> fully deterministic, no RMW hazards.
__global__ __launch_bounds__(256) void k3_sample(
    const float* __restrict__ lab_ws, const float* __restrict__ r_ws,
    float* __restrict__ out_label) {
  const int b = blockIdx.x;
  const float* lab = lab_ws + (size_t)b * A;
  const float* rr = r_ws + (size_t)b * A;
  float* outl = out_label + (size_t)b * A;
  __shared__ unsigned hist[NBIN];
  __shared__ float bnd_r[BCAP];
  __shared__ int bnd_i[BCAP];
  __shared__ int s_total, s_bndn, s_bstar, s_cumb, s_nkeep, s_npos;

  for (int phase = 0; phase < 2; ++phase) {
    const float target = (phase == 0) ? 1.f : 0.f;
    for (int i = threadIdx.x; i < NBIN; i += blockDim.x) hist[i] = 0u;
    if (threadIdx.x == 0) { s_total = 0; s_bndn = 0; }
    __syncthreads();
    for (int a = threadIdx.x; a < A; a += blockDim.x)
      if (lab[a] == target) {
        atomicAdd(&hist[bin_of(rr[a])], 1u);
        atomicAdd(&s_total, 1);
      }
    __syncthreads();
    if (threadIdx.x == 0) {
      const int total = s_total;
      if (phase == 0) s_npos = total;
      const int npk = s_npos < NS / 2 ? s_npos : NS / 2;
      const int nkeep = (phase == 0) ? npk : NS - npk;
      s_nkeep = nkeep;
      if (total <= nkeep) { s_bstar = -1; s_cumb = 0; }
      else {
        int cum = 0, bs = NBIN - 1, cb = 0;
        for (int i = 0; i < NBIN; ++i) {
          const int c = (int)hist[i];
          if (cum + c >= nkeep) { bs = i; cb = cum; break; }
          cum += c;
        }
        s_bstar = bs; s_cumb = cb;
      }
    }
    __syncthreads();
    const int bstar = s_bstar, cumb = s_cumb, nkeep = s_nkeep;
    if (bstar >= 0) {
      for (int a = threadIdx.x; a < A; a += blockDim.x)
        if (lab[a] == target && bin_of(rr[a]) == bstar) {
          const int p = atomicAdd(&s_bndn, 1);
          if (p < BCAP) { bnd_r[p] = rr[a]; bnd_i[p] = a; }
        }
    }
    __syncthreads();
    const int bndn = s_bndn;
    for (int a = threadIdx.x; a < A; a += blockDim.x) {
      if (lab[a] != target) continue;
      bool keepIt = true;
      if (bstar >= 0) {
        const int bi = bin_of(rr[a]);
        if (bi > bstar) keepIt = false;
        else if (bi == bstar) {
          const float ri = rr[a];
          int rank = cumb;                     // stable rank among masked
          if (bndn <= BCAP) {
            for (int j = 0; j < bndn; ++j) {
              const float rj = bnd_r[j];
              rank += (rj < ri) || (rj == ri && bnd_i[j] < a);
            }
          } else {                             // deterministic slow fallback
            for (int j = 0; j < A; ++j)
              if (lab[j] == target && bin_of(rr[j]) == bstar) {
                const float rj = rr[j];
                rank += (rj < ri) || (rj == ri && j < a);
              }
          }
          keepIt = rank < nkeep;
        }
      }
      outl[a] = keepIt ? target : -1.f;
    }
    __syncthreads();
  }
  for (int a = threadIdx.x; a < A; a += blockDim.x) {
    const float l = lab[a];
    if (l != 1.f && l != 0.f) outl[a] = -1.f;
  }
}

__global__ __launch_bounds__(256) void k4_final(
    const float* __restrict__ anchors, const float* __restrict__ gt,
    const int* __restrict__ obj, const int* __restrict__ gt_idx_ws,
    const float* __restrict__ label_final, float* __restrict__ out_boxes,
    float* __restrict__ out_locs, float* __restrict__ out_cls) {
  const int b = blockIdx.y;
  __shared__ float sgt[G * 4];
  __shared__ int sobj[G];
  if (threadIdx.x < G) sobj[threadIdx.x] = obj[b * G + threadIdx.x];
  if (threadIdx.x < 32) {
    tdm_load_dwords_to_lds(gt + (size_t)b * G * 4, lds_offset_of(sgt), G * 4);
    wait_tensorcnt0();
  }
  __syncthreads();
  const int a = blockIdx.x * blockDim.x + threadIdx.x;
  if (a >= A) return;
  const size_t ba = (size_t)b * A + a;
  const float4 av = ((const float4*)anchors)[ba];
  const int gi = gt_idx_ws[ba];
  const float by1 = sgt[gi * 4 + 0], bx1 = sgt[gi * 4 + 1];
  const float by2 = sgt[gi * 4 + 2], bx2 = sgt[gi * 4 + 3];
  float4 bv = {by1, bx1, by2, bx2};
  ((float4*)out_boxes)[ba] = bv;
  const float h = fmaxf(av.z - av.x, 1e-6f), w = fmaxf(av.w - av.y, 1e-6f);
  const float cy = av.x + 0.5f * h, cx = av.y + 0.5f * w;
  const float gh = fmaxf(by2 - by1, 1e-6f), gw = fmaxf(bx2 - bx1, 1e-6f);
  const float gcy = by1 + 0.5f * gh, gcx = bx1 + 0.5f * gw;
  float4 lv = {(gcy - cy) / h, (gcx - cx) / w, logf(gh / h), logf(gw / w)};
  ((float4*)out_locs)[ba] = lv;
  const float lab = label_final[ba];
  out_cls[ba] = (lab == 1.f) ? (float)sobj[gi] : (lab == 0.f ? 0.f : -1.f);
}

// ---------------------------------------------------------------- launcher
extern "C" void kernel_launch(void* const* d_in, const int* in_sizes, int n_in,
                              void* d_out, int out_size, void* d_ws, size_t ws_size,
                              hipStream_t stream) {
  const float* anchors = (const float*)d_in[0];
  const float* gt = (const float*)d_in[1];
  const int* obj = (const int*)d_in[2];

  float* out = (float*)d_out;                 // boxes | locs | label | cls
  float* out_boxes = out;
  float* out_locs = out + (size_t)B * A * 4;
  float* out_label = out + (size_t)2 * B * A * 4;
  float* out_cls = out + (size_t)2 * B * A * 4 + (size_t)B * A;

  char* ws = (char*)d_ws;                     // ~4.33 MB total, 8B-aligned slices
  int* gt_idx_ws = (int*)ws;
  float* r_ws = (float*)(ws + sizeof(int) * (size_t)B * A);
  float* lab_ws = (float*)(ws + 2 * sizeof(int) * (size_t)B * A);
  unsigned long long* best_ws =
      (unsigned long long*)(ws + 3 * sizeof(int) * (size_t)B * A);

  k0_init<<<dim3((B * G + 255) / 256), dim3(256), 0, stream>>>(best_ws);
  k1_iou<<<dim3((A + 255) / 256, B), dim3(256), 0, stream>>>(
      anchors, gt, gt_idx_ws, r_ws, lab_ws, best_ws);
  k2_best<<<dim3((B * G + 255) / 256), dim3(256), 0, stream>>>(best_ws, lab_ws);
  k3_sample<<<dim3(B), dim3(256), 0, stream>>>(lab_ws, r_ws, out_label);
  k4_final<<<dim3((A + 255) / 256, B), dim3(256), 0, stream>>>(
      anchors, gt, obj, gt_idx_ws, out_label, out_boxes, out_locs, out_cls);
}